// WEGAT_Net_19791209300519
// MI455X (gfx1250) — compile-verified
//
#include <hip/hip_runtime.h>
#include <hip/hip_bf16.h>
#include <math.h>

// ---------------- problem constants (match reference) ----------------
#define NN 50000      // nodes
#define EE 1600000    // edges
#define GG 16         // graphs
#define FIN 16        // input node features
#define FE 8          // edge features
#define HD 32         // hidden
#define NEG_SLOPE 0.2f

typedef __attribute__((ext_vector_type(2))) float v2f;
typedef __attribute__((ext_vector_type(8))) float v8f;

// ---------------- helpers ----------------
__device__ __forceinline__ void atomic_max_f32(float* addr, float val) {
  // exact float max via integer atomics (addr initialized to -inf)
  if (val >= 0.0f) atomicMax((int*)addr, __float_as_int(val));
  else             atomicMin((unsigned int*)addr, (unsigned int)__float_as_int(val));
}

__global__ void fill_kernel(float* p, float v, int n) {
  int i = blockIdx.x * blockDim.x + threadIdx.x;
  if (i < n) p[i] = v;
}

// ---------------- node GEMM via V_WMMA_F32_16X16X4_F32 ----------------
// Y[N,32] = act(X)[N,KIN] @ W[KIN,32] + b ; one wave computes a 16x32 tile.
// KIN/RELU are compile-time so the K loop fully unrolls and no cndmask
// selects sit between dependent WMMAs.
template <int KIN, bool RELU>
__global__ void node_gemm_wmma(const float* __restrict__ X,
                               const float* __restrict__ W,
                               const float* __restrict__ bias,
                               float* __restrict__ Y) {
  const int lane = threadIdx.x & 31;
  const int wave = threadIdx.x >> 5;
  const int tile = blockIdx.x * (blockDim.x >> 5) + wave;   // wave-uniform
  if (tile >= NN / 16) return;                              // uniform exit: EXEC stays all-1s

  const int rlo = lane & 15;
  const int hi  = lane >> 4;          // 0: lanes 0-15, 1: lanes 16-31
  const int row = tile * 16 + rlo;

  v8f c0 = {};   // output cols 0..15
  v8f c1 = {};   // output cols 16..31

#pragma unroll
  for (int k0 = 0; k0 < KIN; k0 += 4) {
    // A fragment (ISA 32-bit 16x4 layout): lane holds row M=rlo,
    // VGPR0/1 = K = k0,k0+1 (lanes 0-15) or k0+2,k0+3 (lanes 16-31)
    const int ka = k0 + (hi ? 2 : 0);
    float ax = X[row * KIN + ka];
    float ay = X[row * KIN + ka + 1];
    if (RELU) { ax = fmaxf(ax, 0.0f); ay = fmaxf(ay, 0.0f); }
    v2f a; a.x = ax; a.y = ay;

    // B fragments (row of B = fixed K striped across lanes): col = rlo (+16)
    v2f b0, b1;
    b0.x = W[ ka      * HD + rlo     ];
    b0.y = W[(ka + 1) * HD + rlo     ];
    b1.x = W[ ka      * HD + rlo + 16];
    b1.y = W[(ka + 1) * HD + rlo + 16];

    c0 = __builtin_amdgcn_wmma_f32_16x16x4_f32(false, a, false, b0, (short)0, c0, false, false);
    c1 = __builtin_amdgcn_wmma_f32_16x16x4_f32(false, a, false, b1, (short)0, c1, false, false);
  }

  // C/D layout: VGPR r -> M=r (lanes 0-15) / M=r+8 (lanes 16-31), N=lane%16
  const int rbase = tile * 16 + (hi ? 8 : 0);
#pragma unroll
  for (int r = 0; r < 8; ++r) {
    Y[(rbase + r) * HD + rlo     ] = c0[r] + bias[rlo];
    Y[(rbase + r) * HD + rlo + 16] = c1[r] + bias[rlo + 16];
  }
}

// ---------------- edge MLP: ea_out[E,8] = ea_in @ We(8x8) + be ----------------
__global__ void edge_mlp_kernel(const float* __restrict__ ea_in,
                                const float* __restrict__ We,
                                const float* __restrict__ be,
                                float* __restrict__ ea_out) {
  int e = blockIdx.x * blockDim.x + threadIdx.x;
  if (e >= EE) return;
  const float4* in4 = (const float4*)(ea_in + (size_t)e * FE);
  float4 lo = in4[0], hi = in4[1];
  float a[8] = {lo.x, lo.y, lo.z, lo.w, hi.x, hi.y, hi.z, hi.w};
  float o[8];
#pragma unroll
  for (int j = 0; j < 8; ++j) o[j] = be[j];
#pragma unroll
  for (int i = 0; i < 8; ++i)
#pragma unroll
    for (int j = 0; j < 8; ++j)
      o[j] = fmaf(a[i], We[i * 8 + j], o[j]);
  float4* o4 = (float4*)(ea_out + (size_t)e * FE);
  o4[0] = make_float4(o[0], o[1], o[2], o[3]);
  o4[1] = make_float4(o[4], o[5], o[6], o[7]);
}

// ------- logits: leakyrelu(att . [xn[dst], xn[src], ea2]) ; atomic max into m[dst] -------
__global__ void edge_logits_kernel(const float* __restrict__ xn,
                                   const float* __restrict__ ea2,
                                   const int* __restrict__ src,
                                   const int* __restrict__ dst,
                                   const float* __restrict__ att,
                                   float* __restrict__ logit,
                                   float* __restrict__ m) {
  int e = blockIdx.x * blockDim.x + threadIdx.x;
  if (e >= EE) return;
  const int s = src[e], d = dst[e];
  const float4* xd = (const float4*)(xn + (size_t)d * HD);
  const float4* xs = (const float4*)(xn + (size_t)s * HD);
  const float4* A  = (const float4*)att;   // 72 floats = 18 float4
  float acc = 0.0f;
#pragma unroll
  for (int q = 0; q < 8; ++q) {
    float4 v = xd[q], w = A[q];
    acc += v.x * w.x + v.y * w.y + v.z * w.z + v.w * w.w;
  }
#pragma unroll
  for (int q = 0; q < 8; ++q) {
    float4 v = xs[q], w = A[8 + q];
    acc += v.x * w.x + v.y * w.y + v.z * w.z + v.w * w.w;
  }
  const float4* ev = (const float4*)(ea2 + (size_t)e * FE);
#pragma unroll
  for (int q = 0; q < 2; ++q) {
    float4 v = ev[q], w = A[16 + q];
    acc += v.x * w.x + v.y * w.y + v.z * w.z + v.w * w.w;
  }
  float l = acc > 0.0f ? acc : NEG_SLOPE * acc;
  logit[e] = l;
  atomic_max_f32(&m[d], l);
}

// ------- p = exp(logit - m[dst]) ; atomic sum into s[dst] ; p overwrites logit -------
__global__ void edge_exp_kernel(const int* __restrict__ dst,
                                float* __restrict__ logit,
                                const float* __restrict__ m,
                                float* __restrict__ s) {
  int e = blockIdx.x * blockDim.x + threadIdx.x;
  if (e >= EE) return;
  const int d = dst[e];
  float p = expf(logit[e] - m[d]);
  logit[e] = p;
  atomicAdd(&s[d], p);
}

// ------- scatter: out[dst] += xn[src] * p/(s[dst]+1e-16) ; lane = feature, wave = edge -------
__global__ void edge_scatter_kernel(const float* __restrict__ xn,
                                    const int* __restrict__ src,
                                    const int* __restrict__ dst,
                                    const float* __restrict__ p,
                                    const float* __restrict__ s,
                                    float* __restrict__ out) {
  long long idx = (long long)blockIdx.x * blockDim.x + threadIdx.x;
  int e = (int)(idx >> 5);
  int h = (int)(idx & 31);
  if (e >= EE) return;
  const int d  = dst[e];                    // wave-uniform loads (broadcast)
  const float a = p[e] / (s[d] + 1e-16f);
  const float v = xn[(size_t)src[e] * HD + h] * a;
  atomicAdd(&out[(size_t)d * HD + h], v);
}

// ------- pooling: sums[batch[n]] += x3[n] ; cnt[batch[n]] += 1 -------
__global__ void pool_kernel(const float* __restrict__ x3,
                            const int* __restrict__ batch,
                            float* __restrict__ sums,
                            float* __restrict__ cnt) {
  long long idx = (long long)blockIdx.x * blockDim.x + threadIdx.x;
  int n = (int)(idx >> 5);
  int h = (int)(idx & 31);
  if (n >= NN) return;
  const int b = batch[n];
  atomicAdd(&sums[b * HD + h], x3[(size_t)n * HD + h]);
  if (h == 0) atomicAdd(&cnt[b], 1.0f);
}

// ------- head: out[g] = (sums[g]/max(cnt,1)) . Wlin + blin -------
__global__ void head_kernel(const float* __restrict__ sums,
                            const float* __restrict__ cnt,
                            const float* __restrict__ Wlin,
                            const float* __restrict__ blin,
                            float* __restrict__ out) {
  int g = threadIdx.x;
  if (g >= GG) return;
  const float inv = 1.0f / fmaxf(cnt[g], 1.0f);
  float acc = blin[0];
#pragma unroll
  for (int h = 0; h < HD; ++h) acc += sums[g * HD + h] * inv * Wlin[h];
  out[g] = acc;
}

// ---------------- orchestration ----------------
static inline int cdiv(long long a, int b) { return (int)((a + b - 1) / b); }

template <int KIN, bool RELU>
static void run_layer(const float* x_in,
                      const float* ea_in,
                      const float* Wn, const float* bn,
                      const float* We, const float* be,
                      const float* att,
                      const int* src, const int* dst,
                      float* XN, float* XACC, float* EAO,
                      float* LOG, float* M, float* S,
                      hipStream_t stream) {
  const int TPB = 256;
  // 1. node transform (WMMA fp32)
  node_gemm_wmma<KIN, RELU><<<cdiv(NN / 16, TPB / 32), TPB, 0, stream>>>(x_in, Wn, bn, XN);
  // 2. edge transform
  edge_mlp_kernel<<<cdiv(EE, TPB), TPB, 0, stream>>>(ea_in, We, be, EAO);
  // 3. init m = -inf, s = 0
  fill_kernel<<<cdiv(NN, TPB), TPB, 0, stream>>>(M, -INFINITY, NN);
  fill_kernel<<<cdiv(NN, TPB), TPB, 0, stream>>>(S, 0.0f, NN);
  // 4. logits + segment max
  edge_logits_kernel<<<cdiv(EE, TPB), TPB, 0, stream>>>(XN, EAO, src, dst, att, LOG, M);
  // 5. exp + segment sum
  edge_exp_kernel<<<cdiv(EE, TPB), TPB, 0, stream>>>(dst, LOG, M, S);
  // 6. zero accumulator, weighted scatter
  fill_kernel<<<cdiv((long long)NN * HD, TPB), TPB, 0, stream>>>(XACC, 0.0f, NN * HD);
  edge_scatter_kernel<<<cdiv((long long)EE * 32, TPB), TPB, 0, stream>>>(XN, src, dst, LOG, S, XACC);
}

extern "C" void kernel_launch(void* const* d_in, const int* in_sizes, int n_in,
                              void* d_out, int out_size, void* d_ws, size_t ws_size,
                              hipStream_t stream) {
  const float* x         = (const float*)d_in[0];
  const float* edge_attr = (const float*)d_in[1];
  const int*   eidx      = (const int*)d_in[2];
  const int*   batch     = (const int*)d_in[3];
  const float *Wn1 = (const float*)d_in[4],  *bn1 = (const float*)d_in[5];
  const float *We1 = (const float*)d_in[6],  *be1 = (const float*)d_in[7];
  const float *att1 = (const float*)d_in[8];
  const float *Wn2 = (const float*)d_in[9],  *bn2 = (const float*)d_in[10];
  const float *We2 = (const float*)d_in[11], *be2 = (const float*)d_in[12];
  const float *att2 = (const float*)d_in[13];
  const float *Wn3 = (const float*)d_in[14], *bn3 = (const float*)d_in[15];
  const float *We3 = (const float*)d_in[16], *be3 = (const float*)d_in[17];
  const float *att3 = (const float*)d_in[18];
  const float *Wlin = (const float*)d_in[19], *blin = (const float*)d_in[20];

  const int* src = eidx;        // edge_index[0]
  const int* dst = eidx + EE;   // edge_index[1]

  // workspace carve-up (floats)
  float* ws   = (float*)d_ws;
  float* XN   = ws; ws += (size_t)NN * HD;   // transformed node feats
  float* XACC = ws; ws += (size_t)NN * HD;   // aggregated out / next layer input
  float* EA0  = ws; ws += (size_t)EE * FE;   // edge feats ping
  float* EA1  = ws; ws += (size_t)EE * FE;   // edge feats pong
  float* LOG  = ws; ws += (size_t)EE;        // logits -> p
  float* M    = ws; ws += NN;                // segment max
  float* S    = ws; ws += NN;                // segment sum
  float* POOL = ws; ws += GG * HD;           // pooled sums
  float* CNT  = ws; ws += GG;                // graph node counts

  const int TPB = 256;

  // layer 1: input x[N,16], raw edge_attr -> EA0
  run_layer<FIN, false>(x, edge_attr, Wn1, bn1, We1, be1, att1,
                        src, dst, XN, XACC, EA0, LOG, M, S, stream);
  // layer 2: input relu(XACC), EA0 -> EA1
  run_layer<HD, true>(XACC, EA0, Wn2, bn2, We2, be2, att2,
                      src, dst, XN, XACC, EA1, LOG, M, S, stream);
  // layer 3: input relu(XACC), EA1 -> EA0 (ea3 only used for logits)
  run_layer<HD, true>(XACC, EA1, Wn3, bn3, We3, be3, att3,
                      src, dst, XN, XACC, EA0, LOG, M, S, stream);

  // global mean pool + head (POOL and CNT are contiguous: GG*HD + GG floats)
  fill_kernel<<<cdiv(GG * HD + GG, TPB), TPB, 0, stream>>>(POOL, 0.0f, GG * HD + GG);
  pool_kernel<<<cdiv((long long)NN * 32, TPB), TPB, 0, stream>>>(XACC, batch, POOL, CNT);
  head_kernel<<<1, 32, 0, stream>>>(POOL, CNT, Wlin, blin, (float*)d_out);
}